// ProxyNCALoss_3470333575800
// MI455X (gfx1250) — compile-verified
//
#include <hip/hip_runtime.h>
#include <math.h>

// ProxyNCA loss for MI455X (gfx1250, wave32).
// fp32 GEMM (2048x128 @ 128x100000) via V_WMMA_F32_16X16X4_F32 fused with
// online logsumexp. Proxies (51.2MB) are L2-resident (192MB L2).
// Each block: 32 batch rows (2 WMMA M-tiles reusing every B fragment) and
// half of the class range -> 128 blocks, 3.3GB total L2 reads (was 6.5GB),
// 2 v_wmma per global_load_b64. Normalization folded into the accumulator.

typedef __attribute__((ext_vector_type(2))) float v2f;
typedef __attribute__((ext_vector_type(8))) float v8f;

#define C_CLASSES 100000
#define D_DIM     128
#define B_BATCH   2048
#define PSCALE    10.0f
#define CSPLIT    2
#define CHALF     (C_CLASSES / CSPLIT)   // 50000

__global__ void pnca_zero_out(float* out) { out[0] = 0.0f; }

// One wave per proxy row: pn[c] = { 1/||p_c|| , ||p_c/||p_c||||^2 }
__global__ __launch_bounds__(256) void pnca_proxy_norms(
    const float* __restrict__ proxies, float2* __restrict__ pn)
{
    const int row  = (blockIdx.x * blockDim.x + threadIdx.x) >> 5;
    const int lane = threadIdx.x & 31;
    if (row >= C_CLASSES) return;
    const float4 v = ((const float4*)(proxies + (size_t)row * D_DIM))[lane];
    float s = v.x*v.x + v.y*v.y + v.z*v.z + v.w*v.w;
    for (int o = 16; o >= 1; o >>= 1) s += __shfl_xor(s, o, 32);
    if (lane == 0) {
        const float inv = rsqrtf(s);
        pn[row] = make_float2(inv, s * inv * inv);
    }
}

// One wave per batch row: e_sq[b] and pos[b] = similarity at the true label.
__global__ __launch_bounds__(256) void pnca_row_prep(
    const float* __restrict__ emb, const int* __restrict__ labels,
    const float* __restrict__ proxies, const float2* __restrict__ pn,
    float* __restrict__ e_sq, float* __restrict__ pos)
{
    const int row  = (blockIdx.x * blockDim.x + threadIdx.x) >> 5;
    const int lane = threadIdx.x & 31;
    if (row >= B_BATCH) return;
    const int l = labels[row];
    const float4 e = ((const float4*)(emb     + (size_t)row * D_DIM))[lane];
    const float4 p = ((const float4*)(proxies + (size_t)l   * D_DIM))[lane];
    float es = e.x*e.x + e.y*e.y + e.z*e.z + e.w*e.w;
    float dp = e.x*p.x + e.y*p.y + e.z*p.z + e.w*p.w;
    for (int o = 16; o >= 1; o >>= 1) {
        es += __shfl_xor(es, o, 32);
        dp += __shfl_xor(dp, o, 32);
    }
    if (lane == 0) {
        e_sq[row] = es;
        const float2 pq = pn[l];
        const float sq = es + pq.y - 2.0f * dp * pq.x;
        pos[row] = -PSCALE * sqrtf(fmaxf(sq, 0.0f));
    }
}

// Main fused GEMM + online-logsumexp kernel.
// Grid: 128 blocks x 256 threads (8 waves). Block (g,h): batch rows
// [32g, 32g+32), classes [50000h, 50000h+50000). Each wave holds two A
// fragments (M=0..15 and M=16..31) in registers and per chunk computes two
// 16x16 tiles, reusing each B fragment for 2 WMMAs. Writes per-row (max,sum)
// partials; pnca_finalize merges the CSPLIT partials into the loss.
__global__ __launch_bounds__(256) void pnca_main(
    const float* __restrict__ emb, const float* __restrict__ proxies,
    const float2* __restrict__ pn, const float* __restrict__ e_sq,
    float2* __restrict__ part_ws)
{
    __shared__ float2 part[8][32];           // per-wave (max,sum) per local row

    const int wave = threadIdx.x >> 5;
    const int lane = threadIdx.x & 31;
    const int g    = blockIdx.x >> 1;        // row group (64)
    const int h    = blockIdx.x & 1;         // class half (2)
    const int rb   = g * 32;                 // base batch row
    const int m    = lane & 15;              // A-row / B,C-column slot
    const int kb   = (lane >> 4) << 1;       // K sub-offset: 0 or 2
    const int moff = (lane < 16) ? 0 : 8;    // C/D row offset per VGPR half
    const int cbase = h * CHALF;
    const int cend  = cbase + CHALF;

    // ---- two A fragments (32 rows x 128 K of embeddings) in VGPRs ----
    v2f a0[32], a1[32];
    {
        const float* e0 = emb + (size_t)(rb + m)      * D_DIM + kb;
        const float* e1 = emb + (size_t)(rb + 16 + m) * D_DIM + kb;
#pragma unroll
        for (int s = 0; s < 32; ++s) {
            a0[s] = *(const v2f*)(e0 + 4 * s);
            a1[s] = *(const v2f*)(e1 + 4 * s);
        }
    }

    float es0[8], es1[8];
#pragma unroll
    for (int r = 0; r < 8; ++r) {
        es0[r] = e_sq[rb + r + moff];
        es1[r] = e_sq[rb + 16 + r + moff];
    }

    float mx0[8], sm0[8], mx1[8], sm1[8];
#pragma unroll
    for (int r = 0; r < 8; ++r) {
        mx0[r] = -INFINITY; sm0[r] = 0.0f;
        mx1[r] = -INFINITY; sm1[r] = 0.0f;
    }

    const int NITER = (CHALF + 127) / 128;   // 391
    for (int it = 0; it < NITER; ++it) {
        const int c0  = cbase + it * 128;
        const int n   = c0 + wave * 16 + m;                // my proxy column
        const int nld = (n < C_CLASSES) ? n : (C_CLASSES - 1);
        const float2 pq = pn[nld];                         // {inv_norm, p_sq}
        const float* prow = proxies + (size_t)nld * D_DIM + kb;

        __builtin_prefetch(prow + 128 * D_DIM, 0, 1);      // next chunk

        v8f acc0 = {0.f,0.f,0.f,0.f,0.f,0.f,0.f,0.f};
        v8f acc1 = {0.f,0.f,0.f,0.f,0.f,0.f,0.f,0.f};
#pragma unroll
        for (int s = 0; s < 32; ++s) {
            const v2f b = *(const v2f*)(prow + 4 * s);     // raw proxy data
            acc0 = __builtin_amdgcn_wmma_f32_16x16x4_f32(
                false, a0[s], false, b, (short)0, acc0, false, false);
            acc1 = __builtin_amdgcn_wmma_f32_16x16x4_f32(
                false, a1[s], false, b, (short)0, acc1, false, false);
        }

        if (n < cend) {
            const float t = -2.0f * pq.x;   // fold row normalization here
#pragma unroll
            for (int r = 0; r < 8; ++r) {
                {   // tile 0
                    const float sq  = fmaf(t, acc0[r], es0[r] + pq.y);
                    const float sim = -PSCALE * sqrtf(fmaxf(sq, 0.0f));
                    const float d   = sim - mx0[r];
                    const float e   = __expf(-fabsf(d));
                    sm0[r] = (d > 0.0f) ? fmaf(sm0[r], e, 1.0f) : (sm0[r] + e);
                    mx0[r] = fmaxf(mx0[r], sim);
                }
                {   // tile 1
                    const float sq  = fmaf(t, acc1[r], es1[r] + pq.y);
                    const float sim = -PSCALE * sqrtf(fmaxf(sq, 0.0f));
                    const float d   = sim - mx1[r];
                    const float e   = __expf(-fabsf(d));
                    sm1[r] = (d > 0.0f) ? fmaf(sm1[r], e, 1.0f) : (sm1[r] + e);
                    mx1[r] = fmaxf(mx1[r], sim);
                }
            }
        }
    }

    // ---- reduce (max,sum) across the 16 N-lanes of each half-wave ----
#pragma unroll
    for (int t = 0; t < 16; ++t) {
        const int r = t & 7;
        float m0 = (t < 8) ? mx0[r] : mx1[r];
        float s0 = (t < 8) ? sm0[r] : sm1[r];
        for (int o = 8; o >= 1; o >>= 1) {
            const float m1 = __shfl_xor(m0, o, 32);
            const float s1 = __shfl_xor(s0, o, 32);
            const float nm = fmaxf(m0, m1);
            s0 = s0 * __expf(fmaxf(m0 - nm, -100.0f)) +
                 s1 * __expf(fmaxf(m1 - nm, -100.0f));
            m0 = nm;
        }
        if (m == 0)                          // lanes 0 and 16
            part[wave][((t < 8) ? 0 : 16) + r + moff] = make_float2(m0, s0);
    }
    __syncthreads();

    // ---- wave 0 merges the 8 wave-partials per row, writes global partial ----
    if (wave == 0) {
        float m0 = -INFINITY, s0 = 0.0f;
        for (int w2 = 0; w2 < 8; ++w2) {
            const float2 ps = part[w2][lane];
            const float nm = fmaxf(m0, ps.x);
            s0 = s0   * __expf(fmaxf(m0   - nm, -100.0f)) +
                 ps.y * __expf(fmaxf(ps.x - nm, -100.0f));
            m0 = nm;
        }
        part_ws[h * B_BATCH + rb + lane] = make_float2(m0, s0);
    }
}

// Merge CSPLIT class-range partials per row, finalize weighted loss.
__global__ __launch_bounds__(256) void pnca_finalize(
    const int* __restrict__ labels, const float* __restrict__ cw,
    const float* __restrict__ pos, const float2* __restrict__ part_ws,
    float* __restrict__ out)
{
    const int row  = blockIdx.x * blockDim.x + threadIdx.x;  // 2048 threads
    const int lane = threadIdx.x & 31;
    const float2 p0 = part_ws[row];
    const float2 p1 = part_ws[B_BATCH + row];
    const float nm = fmaxf(p0.x, p1.x);
    const float s  = p0.y * __expf(fmaxf(p0.x - nm, -100.0f)) +
                     p1.y * __expf(fmaxf(p1.x - nm, -100.0f));
    const float lse = nm + __logf(s);
    float contrib = (lse - pos[row]) * cw[labels[row]] * (1.0f / B_BATCH);
    for (int o = 16; o >= 1; o >>= 1) contrib += __shfl_xor(contrib, o, 32);
    if (lane == 0) atomicAdd(out, contrib);
}

extern "C" void kernel_launch(void* const* d_in, const int* in_sizes, int n_in,
                              void* d_out, int out_size, void* d_ws, size_t ws_size,
                              hipStream_t stream)
{
    const float* emb     = (const float*)d_in[0];
    const int*   labels  = (const int*)  d_in[1];
    const float* cw      = (const float*)d_in[2];
    const float* proxies = (const float*)d_in[3];
    float* out = (float*)d_out;

    float*  ws      = (float*)d_ws;
    float2* pn      = (float2*)ws;                               // 2*C floats
    float*  e_sq    = ws + 2 * C_CLASSES;                        // B
    float*  pos     = ws + 2 * C_CLASSES + B_BATCH;              // B
    float2* part_ws = (float2*)(ws + 2 * C_CLASSES + 2 * B_BATCH); // 2*B float2

    pnca_zero_out<<<1, 1, 0, stream>>>(out);
    pnca_proxy_norms<<<(C_CLASSES * 32 + 255) / 256, 256, 0, stream>>>(
        proxies, pn);
    pnca_row_prep<<<(B_BATCH * 32 + 255) / 256, 256, 0, stream>>>(
        emb, labels, proxies, pn, e_sq, pos);
    pnca_main<<<(B_BATCH / 32) * CSPLIT, 256, 0, stream>>>(
        emb, proxies, pn, e_sq, part_ws);
    pnca_finalize<<<B_BATCH / 256, 256, 0, stream>>>(
        labels, cw, pos, part_ws, out);
}